// ConeBackproj3D_29850022707688
// MI455X (gfx1250) — compile-verified
//
#include <hip/hip_runtime.h>
#include <math.h>

// ---- geometry constants (match reference) ----
#define NPROJ 180
#define DET_H 192
#define DET_W 384
#define NZ 96
#define NY 96
#define NX 96
#define PG 5                // projections per WMMA group (3*5 = 15 of 16 A rows)
#define NGRP (NPROJ / PG)   // 36

typedef float v2f __attribute__((ext_vector_type(2)));
typedef float v4f __attribute__((ext_vector_type(4)));
typedef float v8f __attribute__((ext_vector_type(8)));

// Backproject one x-row (96 voxels) per block; 3 waves, each wave = 32 voxels.
__global__ __launch_bounds__(96)
void ConeBackproj3D_kernel(const float* __restrict__ sino,   // [2,180,192,384]
                           const float* __restrict__ mats,   // [180,3,4]
                           float* __restrict__ out)          // [2,96,96,96]
{
    const int y    = blockIdx.x;
    const int z    = blockIdx.y;
    const int b    = blockIdx.z;
    const int tid  = threadIdx.x;        // = x within row, 0..95
    const int wave = tid >> 5;           // 0..2
    const int lane = tid & 31;
    const int half = lane >> 4;          // 0 = lanes 0-15, 1 = lanes 16-31
    const int l16  = lane & 15;
    const int xbase = wave << 5;

    // LDS: projection matrices (reused by all 36 groups) + per-wave D staging.
    __shared__ float s_mats[NPROJ * 12];          // 8640 B
    __shared__ float s_uvw[3 * 2 * 16 * 16];      // 6144 B, [wave][tile][col][row]
    float* myLds = &s_uvw[wave * 512];

    // Cooperative preload of all projection matrices (8.6 KB, L2-resident).
    for (int i = tid; i < NPROJ * 12; i += 96) s_mats[i] = mats[i];
    __syncthreads();

    const float cx = (NX - 1) * 0.5f;
    const float cy = (NY - 1) * 0.5f;
    const float cz = (NZ - 1) * 0.5f;
    const float yw = (float)y - cy;
    const float zw = (float)z - cz;

    // B operand (4x16 f32, K x N). WMMA f32 layout:
    //   VGPR0: lanes 0-15 -> K=0, lanes 16-31 -> K=2
    //   VGPR1: lanes 0-15 -> K=1, lanes 16-31 -> K=3
    // Columns n = voxel within tile; coords (x, y, z, 1).
    v2f bm0, bm1;
    {
        const float x0 = (float)(xbase + l16)      - cx;  // tile 0
        const float x1 = (float)(xbase + 16 + l16) - cx;  // tile 1
        bm0.x = half ? zw   : x0;
        bm0.y = half ? 1.0f : yw;
        bm1.x = half ? zw   : x1;
        bm1.y = half ? 1.0f : yw;
    }

    const float* img_base = sino + (size_t)b * NPROJ * DET_H * DET_W;

    // A-row assignment for this lane:
    // row r = 3*proj_in_group + matrix_row, r = 0..14; row 15 = zeros.
    const int  r      = l16;
    const bool rvalid = (r < 15);
    const int  pr     = r / 3;          // projection within group
    const int  cr     = r - pr * 3;     // row of P (0:u', 1:v', 2:w')
    const int  k0     = half << 1;      // K pair: {0,1} or {2,3}
    const int  abase  = pr * 12 + cr * 4 + k0;   // element offset within group

    float acc = 0.0f;

    for (int g = 0; g < NGRP; ++g) {
        // ---- A operand (16x4 f32) from LDS, branchless (row 15 -> zeros) ----
        const int aidx = rvalid ? (g * (PG * 12) + abase) : 0;   // even -> 8B aligned
        v2f a = *(const v2f*)&s_mats[aidx];                      // ds_load_b64
        if (!rvalid) { a.x = 0.0f; a.y = 0.0f; }                 // v_cndmask x2

        v8f d0 = {0.f, 0.f, 0.f, 0.f, 0.f, 0.f, 0.f, 0.f};
        v8f d1 = {0.f, 0.f, 0.f, 0.f, 0.f, 0.f, 0.f, 0.f};
        // D[r][n] = sum_k P_rows[r][k] * coords[k][n]
        d0 = __builtin_amdgcn_wmma_f32_16x16x4_f32(false, a, false, bm0,
                                                   (short)0, d0, false, false);
        d1 = __builtin_amdgcn_wmma_f32_16x16x4_f32(false, a, false, bm1,
                                                   (short)0, d1, false, false);

        // ---- stage D to LDS transposed: addr = tile*256 + col*16 + row ----
        // Lane holds rows (vgpr + 8*half) of column l16 -> 8 contiguous floats.
        {
            float* p0 = myLds + (0 * 256 + l16 * 16 + half * 8);
            float* p1 = myLds + (1 * 256 + l16 * 16 + half * 8);
            *(v4f*)(p0)     = *((const v4f*)&d0);
            *(v4f*)(p0 + 4) = *((const v4f*)&d0 + 1);
            *(v4f*)(p1)     = *((const v4f*)&d1);
            *(v4f*)(p1 + 4) = *((const v4f*)&d1 + 1);
        }
        asm volatile("s_wait_dscnt 0x0" ::: "memory");

        // ---- consume: lane's voxel is (tile = half, col = l16) ----
        const float* vrow = myLds + (half * 256 + l16 * 16);
        #pragma unroll
        for (int p = 0; p < PG; ++p) {
            const float uh = vrow[3 * p + 0];
            const float vh = vrow[3 * p + 1];
            const float wh = vrow[3 * p + 2];

            const float rw  = __builtin_amdgcn_rcpf(wh);  // v_rcp_f32
            const float u   = uh * rw;
            const float v   = vh * rw;
            const float wgt = rw * rw;                    // FDK 1/w^2 weighting

            const float uf = floorf(u);
            const float vf = floorf(v);
            const int   u0 = (int)uf;
            const int   v0 = (int)vf;
            const float fu = u - uf;
            const float fv = v - vf;
            const int   u1 = u0 + 1;
            const int   v1 = v0 + 1;

            // branchless zero-padded bilinear: clamp indices, load, mask.
            const bool uin0 = (unsigned)u0 < (unsigned)DET_W;
            const bool uin1 = (unsigned)u1 < (unsigned)DET_W;
            const bool vin0 = (unsigned)v0 < (unsigned)DET_H;
            const bool vin1 = (unsigned)v1 < (unsigned)DET_H;
            const int uc0 = min(max(u0, 0), DET_W - 1);
            const int uc1 = min(max(u1, 0), DET_W - 1);
            const int vc0 = min(max(v0, 0), DET_H - 1);
            const int vc1 = min(max(v1, 0), DET_H - 1);

            const float* img = img_base + (size_t)(g * PG + p) * (DET_H * DET_W);
            const unsigned row0 = (unsigned)(vc0 * DET_W);
            const unsigned row1 = (unsigned)(vc1 * DET_W);
            float s00 = img[row0 + (unsigned)uc0];
            float s01 = img[row0 + (unsigned)uc1];
            float s10 = img[row1 + (unsigned)uc0];
            float s11 = img[row1 + (unsigned)uc1];
            s00 = (uin0 & vin0) ? s00 : 0.0f;
            s01 = (uin1 & vin0) ? s01 : 0.0f;
            s10 = (uin0 & vin1) ? s10 : 0.0f;
            s11 = (uin1 & vin1) ? s11 : 0.0f;

            const float gu = 1.0f - fu;
            const float gv = 1.0f - fv;
            float t = s00 * (gu * gv);
            t = fmaf(s01, fu * gv, t);
            t = fmaf(s10, gu * fv, t);
            t = fmaf(s11, fu * fv, t);
            acc = fmaf(wgt, t, acc);
        }
        // Next iteration's LDS stores are in-order after this iteration's
        // ds loads (per-wave LDS ops execute in order), so no extra wait.
    }

    out[(((size_t)b * NZ + z) * NY + y) * NX + (xbase + lane)] = acc;
}

extern "C" void kernel_launch(void* const* d_in, const int* in_sizes, int n_in,
                              void* d_out, int out_size, void* d_ws, size_t ws_size,
                              hipStream_t stream) {
    const float* sino = (const float*)d_in[0];   // [2,180,192,384,1] f32
    const float* mats = (const float*)d_in[1];   // [180,3,4] f32
    float* out = (float*)d_out;                  // [2,96,96,96,1] f32

    dim3 grid(NY, NZ, 2);
    dim3 block(96, 1, 1);
    ConeBackproj3D_kernel<<<grid, block, 0, stream>>>(sino, mats, out);
}